// SchNetEmbedding_50740743635140
// MI455X (gfx1250) — compile-verified
//
#include <hip/hip_runtime.h>
#include <hip/hip_bf16.h>

// ---------------------------------------------------------------------------
// SchNet-style GNN on gfx1250 using V_WMMA_F32_16X16X4_F32 tiles.
// Weights are pre-transposed (and K-padded) into workspace so every B
// fragment is a single global_load_b64; A fragments are hoisted into
// registers and reused across all 8 output column tiles.
// ---------------------------------------------------------------------------

#define HIDF 128
#define NFF  128
#define NGAUSS 50

typedef float v2f __attribute__((ext_vector_type(2)));
typedef float v8f __attribute__((ext_vector_type(8)));

__device__ __forceinline__ v8f wmma4v(v2f a, v2f b, v8f c) {
    // D = A(16x4) * B(4x16) + C, f32 everywhere
    return __builtin_amdgcn_wmma_f32_16x16x4_f32(false, a, false, b, (short)0, c,
                                                 false, false);
}

__device__ __forceinline__ v8f zero8() {
    v8f z;
#pragma unroll
    for (int i = 0; i < 8; ++i) z[i] = 0.0f;
    return z;
}

// shifted softplus: softplus(x) - log(2), numerically stable
__device__ __forceinline__ float sspf(float v) {
    float sp = fmaxf(v, 0.0f) + log1pf(expf(-fabsf(v)));
    return sp - 0.69314718055994531f;
}

#define GSTEP  (10.0f / 49.0f)
#define GCOEFF (-0.5f / (GSTEP * GSTEP))
#define PI_OVER_CUTOFF 0.31415926535897931f

// ---------------------------------------------------------------------------
// dst[c][k] = (k < K) ? src[k][c] : 0     (src: [K][Nc] row-major)
// ---------------------------------------------------------------------------
__global__ void k_transpose_pad(const float* __restrict__ src, float* __restrict__ dst,
                                int K, int Nc, int Kpad) {
    int idx = blockIdx.x * blockDim.x + threadIdx.x;
    if (idx < Nc * Kpad) {
        int c = idx / Kpad, k = idx - c * Kpad;
        dst[idx] = (k < K) ? src[k * Nc + c] : 0.0f;
    }
}

// ---------------------------------------------------------------------------
// h[i,:] = emb[z[i],:]
// ---------------------------------------------------------------------------
__global__ void k_embed(const int* __restrict__ z, const float* __restrict__ emb,
                        float* __restrict__ h, int N) {
    int idx = blockIdx.x * blockDim.x + threadIdx.x;
    if (idx < N * HIDF) {
        int i = idx >> 7, j = idx & 127;
        h[idx] = emb[z[i] * HIDF + j];
    }
}

__global__ void k_zero(float* __restrict__ p, long n) {
    long i = (long)blockIdx.x * blockDim.x + threadIdx.x;
    long stride = (long)gridDim.x * blockDim.x;
    for (; i < n; i += stride) p[i] = 0.0f;
}

// ---------------------------------------------------------------------------
// x = h @ conv_w1   ([N,128] @ [128,128], no bias); bT = conv_w1^T [128][128]
// 128 threads = 4 waves, each wave computes one 16-row tile (8 col-tiles).
// ---------------------------------------------------------------------------
__global__ void __launch_bounds__(128) k_node_xw1(const float* __restrict__ h,
                                                  const float* __restrict__ bT,
                                                  float* __restrict__ x, int N) {
    const int wv   = threadIdx.x >> 5;
    const int lane = threadIdx.x & 31;
    const int n    = lane & 15;
    const int kh   = lane >> 4;
    const int row0 = (blockIdx.x * 4 + wv) * 16;
    const int ar   = min(row0 + n, N - 1);   // clamped A row (stores are guarded)

    const v2f* __restrict__ hv = (const v2f*)h;
    const v2f* __restrict__ bv = (const v2f*)bT;

    v2f af[32];
#pragma unroll
    for (int k = 0; k < 32; ++k) af[k] = hv[(size_t)ar * 64 + 2 * k + kh];

    for (int nt = 0; nt < 8; ++nt) {
        const int col = nt * 16 + n;
        const v2f* __restrict__ bp = bv + (size_t)col * 64 + kh;
        v8f acc = zero8();
#pragma unroll
        for (int k = 0; k < 32; ++k) acc = wmma4v(af[k], bp[2 * k], acc);
#pragma unroll
        for (int i = 0; i < 8; ++i) {
            int m = row0 + i + 8 * kh;
            if (m < N) x[m * NFF + col] = acc[i];
        }
    }
}

// ---------------------------------------------------------------------------
// Fused edge kernel (per layer):
//   d      = |pos[row]-pos[col]|            (recomputed, never stored)
//   attr   = exp(coeff*(d - offset_g)^2)    (LDS, zero-padded K=50 -> 64)
//   t      = ssp(attr @ mlp_w1 + b1)        (WMMA, staged in LDS)
//   W      = (t @ mlp_w2 + b2) * C(d)       (WMMA, kept in accumulators)
//   atomicAdd(agg[col], x[row] * W)         (gather + scatter fused)
// w1T: [128][64] transposed+padded;  w2T: [128][128] transposed.
// ---------------------------------------------------------------------------
__global__ void __launch_bounds__(128) k_edge_conv(
    const int* __restrict__ ei, const float* __restrict__ pos,
    const float* __restrict__ w1T, const float* __restrict__ b1,
    const float* __restrict__ w2T, const float* __restrict__ b2,
    const float* __restrict__ x, float* __restrict__ agg, int N, int E) {

    __shared__ float s_attr[4][16][68];   // padded stride (272B, 8B aligned rows)
    __shared__ float s_hid[4][16][132];   // padded stride (528B, 8B aligned rows)
    __shared__ int   s_src[4][16];
    __shared__ int   s_dst[4][16];
    __shared__ float s_d[4][16];
    __shared__ float s_C[4][16];

    const int wv   = threadIdx.x >> 5;
    const int lane = threadIdx.x & 31;
    const int n    = lane & 15;
    const int kh   = lane >> 4;
    const long e0  = ((long)blockIdx.x * 4 + wv) * 16;

    // ---- phase 1: edge geometry + cutoff ----
    if (lane < 16) {
        long e = e0 + lane;
        int r = 0, c = 0;
        float d = 0.0f, C = 0.0f;
        if (e < E) {
            r = ei[e];
            c = ei[(long)E + e];
            float dx = pos[r * 3 + 0] - pos[c * 3 + 0];
            float dy = pos[r * 3 + 1] - pos[c * 3 + 1];
            float dz = pos[r * 3 + 2] - pos[c * 3 + 2];
            d = sqrtf(dx * dx + dy * dy + dz * dz);
            C = 0.5f * (cosf(d * PI_OVER_CUTOFF) + 1.0f);  // C=0 kills OOB tiles
        }
        s_src[wv][lane] = r; s_dst[wv][lane] = c;
        s_d[wv][lane] = d;   s_C[wv][lane] = C;
    }
    __syncthreads();

    // ---- phase 2: gaussian smearing (K padded to 64 with zeros) ----
    for (int idx = lane; idx < 16 * 64; idx += 32) {
        int m = idx >> 6, g = idx & 63;
        float v = 0.0f;
        if (g < NGAUSS) {
            float t = s_d[wv][m] - (float)g * GSTEP;
            v = expf(GCOEFF * t * t);
        }
        s_attr[wv][m][g] = v;
    }
    __syncthreads();

    // ---- phase 3: t = ssp(attr @ w1 + b1) -> s_hid ----
    {
        const v2f* __restrict__ ap = (const v2f*)&s_attr[wv][n][0];
        const v2f* __restrict__ bv = (const v2f*)w1T;
        v2f af[16];
#pragma unroll
        for (int k = 0; k < 16; ++k) af[k] = ap[2 * k + kh];

        for (int nt = 0; nt < 8; ++nt) {
            const int col = nt * 16 + n;
            const v2f* __restrict__ bp = bv + (size_t)col * 32 + kh;
            v8f acc = zero8();
#pragma unroll
            for (int k = 0; k < 16; ++k) acc = wmma4v(af[k], bp[2 * k], acc);
            float bias = b1[col];
#pragma unroll
            for (int i = 0; i < 8; ++i) {
                int m = i + 8 * kh;
                s_hid[wv][m][col] = sspf(acc[i] + bias);
            }
        }
    }
    __syncthreads();

    // ---- phase 4: W = (t @ w2 + b2) * C ; scatter x[row]*W into agg[col] ----
    {
        const v2f* __restrict__ hp = (const v2f*)&s_hid[wv][n][0];
        const v2f* __restrict__ bv = (const v2f*)w2T;
        v2f af[32];
#pragma unroll
        for (int k = 0; k < 32; ++k) af[k] = hp[2 * k + kh];

        int   xb[8], ab[8];
        float Cm[8];
#pragma unroll
        for (int i = 0; i < 8; ++i) {
            int m = i + 8 * kh;
            xb[i] = s_src[wv][m] * NFF;
            ab[i] = s_dst[wv][m] * NFF;
            Cm[i] = s_C[wv][m];
        }

        for (int nt = 0; nt < 8; ++nt) {
            const int col = nt * 16 + n;
            const v2f* __restrict__ bp = bv + (size_t)col * 64 + kh;
            v8f acc = zero8();
#pragma unroll
            for (int k = 0; k < 32; ++k) acc = wmma4v(af[k], bp[2 * k], acc);
            float bias = b2[col];
#pragma unroll
            for (int i = 0; i < 8; ++i) {
                float W  = (acc[i] + bias) * Cm[i];      // C==0 for padded edges
                float xv = x[xb[i] + col];
                atomicAdd(&agg[ab[i] + col], xv * W);
            }
        }
    }
}

// ---------------------------------------------------------------------------
// h += ssp(agg @ conv_w2 + conv_b2) @ lin_w + lin_b
// w2T: conv_w2^T [128][128];  lwT: lin_w^T [128][128]
// ---------------------------------------------------------------------------
__global__ void __launch_bounds__(128) k_node_out(
    const float* __restrict__ agg, const float* __restrict__ w2T,
    const float* __restrict__ b2c, const float* __restrict__ lwT,
    const float* __restrict__ lb, float* __restrict__ h, int N) {

    __shared__ float s_t[4][16][132];

    const int wv   = threadIdx.x >> 5;
    const int lane = threadIdx.x & 31;
    const int n    = lane & 15;
    const int kh   = lane >> 4;
    const int row0 = (blockIdx.x * 4 + wv) * 16;
    const int ar   = min(row0 + n, N - 1);

    // t = ssp(agg @ w2c + b2c) -> LDS
    {
        const v2f* __restrict__ av = (const v2f*)agg;
        const v2f* __restrict__ bv = (const v2f*)w2T;
        v2f af[32];
#pragma unroll
        for (int k = 0; k < 32; ++k) af[k] = av[(size_t)ar * 64 + 2 * k + kh];

        for (int nt = 0; nt < 8; ++nt) {
            const int col = nt * 16 + n;
            const v2f* __restrict__ bp = bv + (size_t)col * 64 + kh;
            v8f acc = zero8();
#pragma unroll
            for (int k = 0; k < 32; ++k) acc = wmma4v(af[k], bp[2 * k], acc);
            float bias = b2c[col];
#pragma unroll
            for (int i = 0; i < 8; ++i) {
                int m = i + 8 * kh;
                s_t[wv][m][col] = sspf(acc[i] + bias);
            }
        }
    }
    __syncthreads();

    // h += t @ lw + lb
    {
        const v2f* __restrict__ tp = (const v2f*)&s_t[wv][n][0];
        const v2f* __restrict__ bv = (const v2f*)lwT;
        v2f af[32];
#pragma unroll
        for (int k = 0; k < 32; ++k) af[k] = tp[2 * k + kh];

        for (int nt = 0; nt < 8; ++nt) {
            const int col = nt * 16 + n;
            const v2f* __restrict__ bp = bv + (size_t)col * 64 + kh;
            v8f acc = zero8();
#pragma unroll
            for (int k = 0; k < 32; ++k) acc = wmma4v(af[k], bp[2 * k], acc);
            float bias = lb[col];
#pragma unroll
            for (int i = 0; i < 8; ++i) {
                int m = row0 + i + 8 * kh;
                if (m < N) h[m * HIDF + col] += acc[i] + bias;
            }
        }
    }
}

// ---------------------------------------------------------------------------
// out = ssp(h @ lin1_w + lin1_b)   ([N,128] @ [128,64]);  bT: lin1_w^T [64][128]
// ---------------------------------------------------------------------------
__global__ void __launch_bounds__(128) k_final(const float* __restrict__ h,
                                               const float* __restrict__ bT,
                                               const float* __restrict__ b,
                                               float* __restrict__ out, int N) {
    const int wv   = threadIdx.x >> 5;
    const int lane = threadIdx.x & 31;
    const int n    = lane & 15;
    const int kh   = lane >> 4;
    const int row0 = (blockIdx.x * 4 + wv) * 16;
    const int ar   = min(row0 + n, N - 1);
    const int NO   = HIDF / 2;

    const v2f* __restrict__ hv = (const v2f*)h;
    const v2f* __restrict__ bv = (const v2f*)bT;

    v2f af[32];
#pragma unroll
    for (int k = 0; k < 32; ++k) af[k] = hv[(size_t)ar * 64 + 2 * k + kh];

    for (int nt = 0; nt < 4; ++nt) {
        const int col = nt * 16 + n;
        const v2f* __restrict__ bp = bv + (size_t)col * 64 + kh;
        v8f acc = zero8();
#pragma unroll
        for (int k = 0; k < 32; ++k) acc = wmma4v(af[k], bp[2 * k], acc);
        float bias = b[col];
#pragma unroll
        for (int i = 0; i < 8; ++i) {
            int m = row0 + i + 8 * kh;
            if (m < N) out[m * NO + col] = sspf(acc[i] + bias);
        }
    }
}

// ---------------------------------------------------------------------------
extern "C" void kernel_launch(void* const* d_in, const int* in_sizes, int n_in,
                              void* d_out, int out_size, void* d_ws, size_t ws_size,
                              hipStream_t stream) {
    const int*   z       = (const int*)d_in[0];
    const float* pos     = (const float*)d_in[1];
    /* d_in[2] = batch (unused) */
    const int*   ei      = (const int*)d_in[3];
    const float* emb     = (const float*)d_in[4];
    const float* mlp_w1  = (const float*)d_in[5];
    const float* mlp_b1  = (const float*)d_in[6];
    const float* mlp_w2  = (const float*)d_in[7];
    const float* mlp_b2  = (const float*)d_in[8];
    const float* conv_w1 = (const float*)d_in[9];
    const float* conv_w2 = (const float*)d_in[10];
    const float* conv_b2 = (const float*)d_in[11];
    const float* lin_w   = (const float*)d_in[12];
    const float* lin_b   = (const float*)d_in[13];
    const float* lin1_w  = (const float*)d_in[14];
    const float* lin1_b  = (const float*)d_in[15];
    float* out = (float*)d_out;

    const int N = in_sizes[0];
    const int E = in_sizes[3] / 2;
    const int L = 3;

    float* h    = (float*)d_ws;                  // [N,128]
    float* x    = h   + (size_t)N * HIDF;        // [N,128]
    float* agg  = x   + (size_t)N * HIDF;        // [N,128]
    float* w1T  = agg + (size_t)N * HIDF;        // 3 x [128][64]  (padded K 50->64)
    float* w2T  = w1T + (size_t)L * NFF * 64;    // 3 x [128][128]
    float* cw1T = w2T + (size_t)L * NFF * NFF;   // 3 x [128][128]
    float* cw2T = cw1T + (size_t)L * NFF * NFF;  // 3 x [128][128]
    float* lwT  = cw2T + (size_t)L * NFF * NFF;  // 3 x [128][128]
    float* l1T  = lwT + (size_t)L * NFF * NFF;   // [64][128]

    // ---- one-time weight transposes (graph-captured; deterministic) ----
    for (int l = 0; l < L; ++l) {
        k_transpose_pad<<<(NFF * 64 + 255) / 256, 256, 0, stream>>>(
            mlp_w1 + (size_t)l * NGAUSS * NFF, w1T + (size_t)l * NFF * 64,
            NGAUSS, NFF, 64);
        k_transpose_pad<<<(NFF * NFF + 255) / 256, 256, 0, stream>>>(
            mlp_w2 + (size_t)l * NFF * NFF, w2T + (size_t)l * NFF * NFF,
            NFF, NFF, NFF);
        k_transpose_pad<<<(NFF * NFF + 255) / 256, 256, 0, stream>>>(
            conv_w1 + (size_t)l * HIDF * NFF, cw1T + (size_t)l * NFF * NFF,
            HIDF, NFF, HIDF);
        k_transpose_pad<<<(NFF * NFF + 255) / 256, 256, 0, stream>>>(
            conv_w2 + (size_t)l * NFF * HIDF, cw2T + (size_t)l * NFF * NFF,
            NFF, HIDF, NFF);
        k_transpose_pad<<<(NFF * NFF + 255) / 256, 256, 0, stream>>>(
            lin_w + (size_t)l * HIDF * HIDF, lwT + (size_t)l * NFF * NFF,
            HIDF, HIDF, HIDF);
    }
    k_transpose_pad<<<(64 * HIDF + 255) / 256, 256, 0, stream>>>(
        lin1_w, l1T, HIDF, 64, HIDF);

    const int node_blocks = ((N + 15) / 16 + 3) / 4;
    const int edge_blocks = (int)((((long)E + 15) / 16 + 3) / 4);

    k_embed<<<(N * HIDF + 255) / 256, 256, 0, stream>>>(z, emb, h, N);

    for (int l = 0; l < L; ++l) {
        k_zero<<<1024, 256, 0, stream>>>(agg, (long)N * HIDF);
        k_node_xw1<<<node_blocks, 128, 0, stream>>>(
            h, cw1T + (size_t)l * NFF * NFF, x, N);
        k_edge_conv<<<edge_blocks, 128, 0, stream>>>(
            ei, pos,
            w1T + (size_t)l * NFF * 64, mlp_b1 + (size_t)l * NFF,
            w2T + (size_t)l * NFF * NFF, mlp_b2 + (size_t)l * NFF,
            x, agg, N, E);
        k_node_out<<<node_blocks, 128, 0, stream>>>(
            agg, cw2T + (size_t)l * NFF * NFF, conv_b2 + (size_t)l * HIDF,
            lwT + (size_t)l * NFF * NFF, lin_b + (size_t)l * HIDF, h, N);
    }

    k_final<<<node_blocks, 128, 0, stream>>>(h, l1T, lin1_b, out, N);
}